// CRFLayer_32392643346443
// MI455X (gfx1250) — compile-verified
//
#include <hip/hip_runtime.h>

static constexpr int CB = 1024;   // batch
static constexpr int CT = 512;    // time
static constexpr int CL = 32;     // labels

typedef __attribute__((ext_vector_type(2))) float v2f;
typedef __attribute__((ext_vector_type(8))) float v8f;

// ---------------------------------------------------------------------------
// Kernel 1: gold path score (unary + binary), one wave32 per sequence.
// Coalesced label reads (lane-strided over t), wave butterfly reduction.
// ---------------------------------------------------------------------------
__global__ __launch_bounds__(128) void crf_path_score(
    const float* __restrict__ logits, const float* __restrict__ trans,
    const int* __restrict__ labels, const int* __restrict__ seq_lens,
    float* __restrict__ out)
{
  const int wid  = (blockIdx.x * blockDim.x + threadIdx.x) >> 5;   // sequence
  const int lane = threadIdx.x & 31;
  int len = seq_lens[wid];
  len = len < 1 ? 1 : (len > CT ? CT : len);
  const int*   lb = labels + (size_t)wid * CT;
  const float* lg = logits + (size_t)wid * CT * CL;
  float acc = 0.0f;
  for (int t = lane; t < CT; t += 32) {
    if (t < len) {
      int c = lb[t];
      acc += lg[(size_t)t * CL + c];                 // unary
      if (t >= 1) acc += trans[lb[t - 1] * CL + c];  // binary
    }
  }
  #pragma unroll
  for (int off = 16; off >= 1; off >>= 1) acc += __shfl_xor(acc, off, 32);
  if (lane == 0) out[wid] = acc;
}

// ---------------------------------------------------------------------------
// Kernel 2: log-partition via forward algorithm in scaled-probability space.
// One wave32 per tile of 16 sequences. Per step:
//   q[16x32] = p[16x32] @ exp(trans)[32x32]   -> 16x V_WMMA_F32_16X16X4_F32
//   q *= exp(logit_t)                          (C layout)
//   transpose C->A layout via padded LDS, mask by seq_len, rescale.
//
// Assumed VGPR layouts for V_WMMA_F32_16X16X4_F32 (ISA 7.12.2, wave32):
//   A (16x4, MxK): lane l -> M = l%16; VGPR v holds K = v + 2*(l/16)
//   B (4x16, KxN): lane l -> N = l%16; VGPR v holds K = v + 2*(l/16)
//   C/D (16x16):   lane l -> N = l%16; VGPR g holds M = g + 8*(l/16)
// ---------------------------------------------------------------------------
__global__ __launch_bounds__(128, 1) void crf_log_norm(
    const float* __restrict__ logits, const float* __restrict__ trans,
    const int* __restrict__ seq_lens, float* __restrict__ out)
{
  __shared__ float lds[4 * 16 * 34];     // 4 waves * 16 rows * stride 34 (bank-conflict-free)
  const int lane = threadIdx.x & 31;
  const int wv   = threadIdx.x >> 5;
  const int tile = blockIdx.x * 4 + wv;  // 0..63, 16 sequences each
  const int r = lane & 15;
  const int h = lane >> 4;

  // B operand: exp(trans) in register, B layout. 8 K-chunks x 2 N-tiles.
  v2f Bop[8][2];
  #pragma unroll
  for (int kc = 0; kc < 8; ++kc)
    #pragma unroll
    for (int nt = 0; nt < 2; ++nt)
      #pragma unroll
      for (int v = 0; v < 2; ++v) {
        const int i = 4 * kc + v + 2 * h;   // K: source label
        const int j = 16 * nt + r;          // N: dest label
        Bop[kc][nt][v] = __expf(trans[i * CL + j]);
      }

  const int seqA = tile * 16 + r;           // sequence owned by this lane in A layout
  int lenA = seq_lens[seqA];
  lenA = lenA < 1 ? 1 : (lenA > CT ? CT : lenA);
  int tmax = lenA;                          // max len within the 16-seq tile
  #pragma unroll
  for (int off = 1; off <= 8; off <<= 1) {
    int o = __shfl_xor(tmax, off, 32);
    tmax = o > tmax ? o : tmax;
  }
  tmax = __builtin_amdgcn_readfirstlane(tmax);   // uniform loop bound (keeps EXEC all-1s)

  // p = exp(alpha0) = exp(logits[:,0,:]) loaded directly in A layout; c = 0.
  v2f pA[8];
  #pragma unroll
  for (int kc = 0; kc < 8; ++kc)
    #pragma unroll
    for (int v = 0; v < 2; ++v) {
      const int k = 4 * kc + v + 2 * h;
      pA[kc][v] = __expf(logits[(size_t)seqA * CT * CL + k]);
    }
  float cA = 0.0f;                          // running log-scale, per sequence (A layout)

  float* Lw = lds + wv * (16 * 34);         // wave-private LDS region
  const int stb = h * 8 * 34 + r;           // C-layout store base (row = g+8h, col = r+16nt)
  const int ldb = r * 34 + 2 * h;           // A-layout load base  (row = r, col = 4kc+v+2h)

  for (int t = 1; t < tmax; ++t) {
    // ---- q = p @ exp(trans): 16 fp32 WMMAs, accumulate over 8 K-chunks
    v8f q0 = {0,0,0,0,0,0,0,0};
    v8f q1 = {0,0,0,0,0,0,0,0};
    #pragma unroll
    for (int kc = 0; kc < 8; ++kc) {
      q0 = __builtin_amdgcn_wmma_f32_16x16x4_f32(false, pA[kc], false, Bop[kc][0],
                                                 (short)0, q0, false, false);
      q1 = __builtin_amdgcn_wmma_f32_16x16x4_f32(false, pA[kc], false, Bop[kc][1],
                                                 (short)0, q1, false, false);
    }

    // ---- q *= exp(emission_t) in C layout (one addr VGPR, imm24 offsets)
    const float* lgb = logits + ((size_t)(tile * 16 + 8 * h) * CT + t) * CL + r;
    #pragma unroll
    for (int g = 0; g < 8; ++g) {
      q0[g] *= __expf(lgb[(size_t)g * CT * CL]);
      q1[g] *= __expf(lgb[(size_t)g * CT * CL + 16]);
    }
    // prefetch next timestep's 128B emission row per sequence
    if (t + 1 < tmax)
      __builtin_prefetch(logits + ((size_t)seqA * CT + t + 1) * CL, 0, 3);

    // ---- C layout -> A layout cross-lane transpose through padded LDS
    #pragma unroll
    for (int g = 0; g < 8; ++g) {
      Lw[stb + g * 34]      = q0[g];
      Lw[stb + g * 34 + 16] = q1[g];
    }
    asm volatile("" ::: "memory");   // keep compiler from reordering; DS is in-order in HW
    float qa[16];
    #pragma unroll
    for (int kc = 0; kc < 8; ++kc) {
      qa[2 * kc]     = Lw[ldb + 4 * kc];
      qa[2 * kc + 1] = Lw[ldb + 4 * kc + 1];
    }

    // ---- sequence-length mask: freeze p (and thus alpha = log p + c) past len
    const bool act = (t < lenA);
    #pragma unroll
    for (int kc = 0; kc < 8; ++kc) {
      pA[kc][0] = act ? qa[2 * kc]     : pA[kc][0];
      pA[kc][1] = act ? qa[2 * kc + 1] : pA[kc][1];
    }

    // ---- periodic per-sequence renormalization (fp32 range safety)
    if ((t & 3) == 3) {
      float m = pA[0][0];
      #pragma unroll
      for (int kc = 0; kc < 8; ++kc) {
        m = fmaxf(m, pA[kc][0]);
        m = fmaxf(m, pA[kc][1]);
      }
      m = fmaxf(m, __shfl_xor(m, 16, 32));   // combine both label halves of the row
      const float inv = __builtin_amdgcn_rcpf(m);
      cA += __logf(m);                        // log p + c invariant preserved
      #pragma unroll
      for (int kc = 0; kc < 8; ++kc) { pA[kc][0] *= inv; pA[kc][1] *= inv; }
    }
  }

  // log_norm = c + log(sum_j p_j)
  float s = 0.0f;
  #pragma unroll
  for (int kc = 0; kc < 8; ++kc) s += pA[kc][0] + pA[kc][1];
  s += __shfl_xor(s, 16, 32);
  const float log_norm = cA + __logf(s);
  if (lane < 16) out[seqA] = out[seqA] - log_norm;   // path score written by kernel 1
}

// ---------------------------------------------------------------------------
extern "C" void kernel_launch(void* const* d_in, const int* in_sizes, int n_in,
                              void* d_out, int out_size, void* d_ws, size_t ws_size,
                              hipStream_t stream) {
  (void)in_sizes; (void)n_in; (void)out_size; (void)d_ws; (void)ws_size;
  const float* logits   = (const float*)d_in[0];
  const float* trans    = (const float*)d_in[1];
  const int*   labels   = (const int*)d_in[2];
  const int*   seq_lens = (const int*)d_in[3];
  float* out = (float*)d_out;

  // 1024 waves: gold path score -> d_out
  crf_path_score<<<dim3(CB / 4), dim3(128), 0, stream>>>(logits, trans, labels, seq_lens, out);
  // 64 waves (16 seqs each): d_out -= log_norm
  crf_log_norm<<<dim3(CB / 64), dim3(128), 0, stream>>>(logits, trans, seq_lens, out);
}